// GNNEncoder_17308718203031
// MI455X (gfx1250) — compile-verified
//
#include <hip/hip_runtime.h>

#define N_NODES 50000
#define N_EDGES 600000
#define DIM 128

typedef float v2f  __attribute__((ext_vector_type(2)));
typedef float v8f  __attribute__((ext_vector_type(8)));
typedef unsigned int u32x4 __attribute__((ext_vector_type(4)));
typedef int   i32x4 __attribute__((ext_vector_type(4)));
typedef int   i32x8 __attribute__((ext_vector_type(8)));

// ---------------------------------------------------------------------------
// init helpers
// ---------------------------------------------------------------------------
__global__ void zero_int(int* __restrict__ p, int n) {
    int i = blockIdx.x * blockDim.x + threadIdx.x;
    if (i < n) p[i] = 0;
}

__global__ void zero_f32(float* __restrict__ p, int n) {
    int i = blockIdx.x * blockDim.x + threadIdx.x;
    if (i < n) p[i] = 0.0f;
}

// histogram of destination cols (CSR build) + source-row degrees (for dinv)
__global__ void hist_kernel(const int* __restrict__ row, const int* __restrict__ col,
                            int* __restrict__ cnt, float* __restrict__ deg) {
    int e = blockIdx.x * blockDim.x + threadIdx.x;
    if (e < N_EDGES) {
        atomicAdd(&cnt[col[e]], 1);
        unsafeAtomicAdd(&deg[row[e]], 1.0f);
    }
}

__global__ void dinv_kernel(float* __restrict__ deg) {
    int i = blockIdx.x * blockDim.x + threadIdx.x;
    if (i < N_NODES) {
        float d = deg[i];
        d = d < 1.0f ? 1.0f : d;
        deg[i] = rsqrtf(d);          // deg^-0.5
    }
}

// single-block exclusive scan over cnt[0..N) -> start[0..N], cursor copy
__global__ __launch_bounds__(1024) void scan_kernel(const int* __restrict__ cnt,
                                                    int* __restrict__ start,
                                                    int* __restrict__ cursor) {
    __shared__ int sums[1024];
    const int t  = threadIdx.x;
    const int CH = (N_NODES + 1023) / 1024;   // 49 elements per thread
    const int base = t * CH;

    int local = 0;
    for (int i = 0; i < CH; ++i) {
        int idx = base + i;
        if (idx < N_NODES) local += cnt[idx];
    }
    sums[t] = local;
    __syncthreads();
    for (int off = 1; off < 1024; off <<= 1) {   // Hillis-Steele inclusive scan
        int v = (t >= off) ? sums[t - off] : 0;
        __syncthreads();
        sums[t] += v;
        __syncthreads();
    }
    int excl = (t == 0) ? 0 : sums[t - 1];
    for (int i = 0; i < CH; ++i) {
        int idx = base + i;
        if (idx < N_NODES) {
            start[idx]  = excl;
            cursor[idx] = excl;
            excl += cnt[idx];
        }
    }
    if (t == 1023) start[N_NODES] = sums[1023];
}

// bucket edges by destination col
__global__ void scatter_kernel(const int* __restrict__ col,
                               int* __restrict__ cursor, int* __restrict__ eidx) {
    int e = blockIdx.x * blockDim.x + threadIdx.x;
    if (e < N_EDGES) {
        int p = atomicAdd(&cursor[col[e]], 1);
        eidx[p] = e;
    }
}

// ---------------------------------------------------------------------------
// h_out = (relu_in ? relu(h_in) : h_in) @ Wt^T + bt
// V_WMMA_F32_16X16X4_F32, wave32. One wave -> one 16x16 tile; block = 8 waves
// = all 8 N-tiles of a 16-row stripe. The 16x128 A stripe is DMA'd once per
// block into LDS by the Tensor Data Mover, then read as ds_load_b64 fragments.
// ---------------------------------------------------------------------------
__global__ __launch_bounds__(256) void gemm_kernel(const float* __restrict__ H,
                                                   const float* __restrict__ Wt,
                                                   const float* __restrict__ bt,
                                                   float* __restrict__ Ht,
                                                   int relu_in) {
    __shared__ float lA[16 * DIM];            // 8 KB, LDS offset 0

    const int lane = threadIdx.x & 31;
    const int wave = threadIdx.x >> 5;
    const int half = lane >> 4;               // 0: lanes 0-15, 1: lanes 16-31
    const int r    = lane & 15;
    const int m0   = blockIdx.x * 16;         // 3125 blocks * 16 = 50000 rows
    const int n0   = wave * 16;               // 8 waves * 16 = 128 cols

    // The TDM intrinsic's LDS write is invisible to alias analysis; without a
    // visible (potential) store the compiler folds all lA loads to undef.
    // relu_in is only ever 0 or 1 at runtime, so this store never executes,
    // but it forces the ds_load_b64 fragment reads to stay real.
    if (relu_in > 1) lA[threadIdx.x] = 0.0f;

    if (wave == 0) {
        // Tensor DMA descriptor (D#): 2D tile 128 x 16 of 4-byte elements
        unsigned long long ga = (unsigned long long)(uintptr_t)(H + (size_t)m0 * DIM);
        u32x4 g0;
        g0[0] = 1u;                                            // count=1 (valid, user)
        g0[1] = 0u;                                            // lds_addr = 0 (lA)
        g0[2] = (unsigned)(ga & 0xffffffffu);                  // global_addr[31:0]
        g0[3] = (unsigned)((ga >> 32) & 0x1ffffffu) | (2u << 30); // [56:32] | type=2
        i32x8 g1;
        g1[0] = (int)(2u << 16);        // data_size = 4B; no multicast/iterate/pad
        g1[1] = (int)(128u << 16);      // tensor_dim0 = 128  (bits 79:48, low half)
        g1[2] = (int)(16u  << 16);      // tensor_dim1 = 16   (bits 111:80, low half)
        g1[3] = (int)(128u << 16);      // tile_dim0   = 128  (bits 127:112)
        g1[4] = 16;                     // tile_dim1 = 16, tile_dim2 = 0
        g1[5] = 128;                    // tensor_dim0_stride = 128
        g1[6] = 0;
        g1[7] = 0;
        i32x4 z4 = {0, 0, 0, 0};                 // groups 2/3 unused (2D tensor)
        i32x8 z8 = {0, 0, 0, 0, 0, 0, 0, 0};     // extra group (clang-23 6-arg form)
        __builtin_amdgcn_tensor_load_to_lds(g0, g1, z4, z4, z8, 0);
        __builtin_amdgcn_s_wait_tensorcnt(0);
    }
    __syncthreads();

    // A frag from LDS: row M = r, K pair selected by lane half (ds_load_b64)
    // B frag from global: column N = r; B[k][n] = Wt[n][k]
    const float* arow = lA + (size_t)r * DIM + half * 2;
    const float* brow = Wt + (size_t)(n0 + r) * DIM + half * 2;

    v8f c = {};
#pragma unroll
    for (int kk = 0; kk < DIM; kk += 4) {
        v2f a = *(const v2f*)(arow + kk);
        if (relu_in) {                        // fused ReLU of previous layer
            a.x = fmaxf(a.x, 0.0f);
            a.y = fmaxf(a.y, 0.0f);
        }
        v2f b = *(const v2f*)(brow + kk);
        c = __builtin_amdgcn_wmma_f32_16x16x4_f32(false, a, false, b,
                                                  (short)0, c, false, false);
    }

    const float bias = bt[n0 + r];
#pragma unroll
    for (int i = 0; i < 8; ++i) {
        // C/D layout: VGPR i -> M = i (lanes 0-15) or i+8 (lanes 16-31); N = r
        int m = m0 + i + half * 8;
        Ht[(size_t)m * DIM + n0 + r] = c[i] + bias;
    }
}

// ---------------------------------------------------------------------------
// Per-edge gate: norm[e] = tanh(h[row]ꞏWg + bg) * dinv[row] * dinv[col]
// One wave per edge; float4 gather per lane + butterfly reduction.
// ---------------------------------------------------------------------------
__global__ __launch_bounds__(256) void gate_kernel(const float* __restrict__ Ht,
                                                   const int* __restrict__ row,
                                                   const int* __restrict__ col,
                                                   const float* __restrict__ Wg,
                                                   const float* __restrict__ bg,
                                                   const float* __restrict__ dinv,
                                                   float* __restrict__ norm) {
    const int lane = threadIdx.x & 31;
    const int e    = blockIdx.x * 8 + (threadIdx.x >> 5);
    if (e >= N_EDGES) return;

    const int r = row[e];
    const int c = col[e];

    const float4 hv = *(const float4*)(Ht + (size_t)r * DIM + lane * 4);
    const float4 wg = *(const float4*)(Wg + lane * 4);

    float p = hv.x * wg.x + hv.y * wg.y + hv.z * wg.z + hv.w * wg.w;
#pragma unroll
    for (int off = 16; off > 0; off >>= 1)
        p += __shfl_xor(p, off, 32);                 // wave32 butterfly

    if (lane == 0)
        norm[e] = tanhf(p + bg[0]) * dinv[r] * dinv[c];
}

// ---------------------------------------------------------------------------
// Per-node aggregation over CSR(col): out[n] = ht[n] + sum_e norm[e]*ht[row[e]]
// One wave per node, 4 features/lane in registers -> ZERO fp32 atomics.
// ---------------------------------------------------------------------------
__global__ __launch_bounds__(256) void aggregate_kernel(const float* __restrict__ Ht,
                                                        const int* __restrict__ row,
                                                        const int* __restrict__ start,
                                                        const int* __restrict__ eidx,
                                                        const float* __restrict__ norm,
                                                        float* __restrict__ out,
                                                        int do_relu) {
    const int lane = threadIdx.x & 31;
    const int n    = blockIdx.x * 8 + (threadIdx.x >> 5);
    if (n >= N_NODES) return;

    float4 acc = *(const float4*)(Ht + (size_t)n * DIM + lane * 4);   // residual
    const int pEnd = start[n + 1];
    for (int p = start[n]; p < pEnd; ++p) {
        const int   e = eidx[p];
        const int   r = row[e];
        const float w = norm[e];
        const float4 hv = *(const float4*)(Ht + (size_t)r * DIM + lane * 4);
        acc.x += w * hv.x;  acc.y += w * hv.y;
        acc.z += w * hv.z;  acc.w += w * hv.w;
    }
    if (do_relu) {
        acc.x = fmaxf(acc.x, 0.0f);  acc.y = fmaxf(acc.y, 0.0f);
        acc.z = fmaxf(acc.z, 0.0f);  acc.w = fmaxf(acc.w, 0.0f);
    }
    *(float4*)(out + (size_t)n * DIM + lane * 4) = acc;
}

// ---------------------------------------------------------------------------
extern "C" void kernel_launch(void* const* d_in, const int* in_sizes, int n_in,
                              void* d_out, int out_size, void* d_ws, size_t ws_size,
                              hipStream_t stream) {
    const float* x   = (const float*)d_in[0];           // [N, 128]
    const int*   ei  = (const int*)d_in[1];             // [2, E]
    const int*   row = ei;
    const int*   col = ei + N_EDGES;
    const float* Wt  = (const float*)d_in[2];           // [2,128,128]
    const float* bt  = (const float*)d_in[3];           // [2,128]
    const float* Wg  = (const float*)d_in[4];           // [2,128]
    const float* bg  = (const float*)d_in[5];           // [2]
    float*       out = (float*)d_out;                   // [N, 128]

    // workspace layout
    float* ht     = (float*)d_ws;                       // 25.6 MB
    float* h1     = ht   + (size_t)N_NODES * DIM;       // 25.6 MB
    float* normE  = h1   + (size_t)N_NODES * DIM;       // 2.4 MB
    float* dinv   = normE + N_EDGES;                    // 0.2 MB
    int*   cnt    = (int*)(dinv + N_NODES);             // 0.2 MB
    int*   start  = cnt   + N_NODES;                    // 0.2 MB (N+1)
    int*   cursor = start + N_NODES + 1;                // 0.2 MB
    int*   eidx   = cursor + N_NODES;                   // 2.4 MB

    const int B = 256;
    const int gN  = (N_NODES + B - 1) / B;
    const int gE  = (N_EDGES + B - 1) / B;
    const int gEw = (N_EDGES + 7) / 8;                  // wave-per-edge grids
    const int gNw = (N_NODES + 7) / 8;                  // wave-per-node grids

    // ---- graph structure (depends only on edge_index; built once) ----
    zero_int<<<gN, B, 0, stream>>>(cnt, N_NODES);
    zero_f32<<<gN, B, 0, stream>>>(dinv, N_NODES);
    hist_kernel<<<gE, B, 0, stream>>>(row, col, cnt, dinv);
    scan_kernel<<<1, 1024, 0, stream>>>(cnt, start, cursor);
    scatter_kernel<<<gE, B, 0, stream>>>(col, cursor, eidx);
    dinv_kernel<<<gN, B, 0, stream>>>(dinv);

    // ---- layer 0 ----
    gemm_kernel<<<N_NODES / 16, B, 0, stream>>>(x, Wt, bt, ht, 0);
    gate_kernel<<<gEw, B, 0, stream>>>(ht, row, col, Wg, bg, dinv, normE);
    aggregate_kernel<<<gNw, B, 0, stream>>>(ht, row, start, eidx, normE, h1, 0);

    // ---- layer 1 (ReLU of layer 0 fused into A-load; final ReLU fused) ----
    gemm_kernel<<<N_NODES / 16, B, 0, stream>>>(h1, Wt + DIM * DIM, bt + DIM, ht, 1);
    gate_kernel<<<gEw, B, 0, stream>>>(ht, row, col, Wg + DIM, bg + 1, dinv, normE);
    aggregate_kernel<<<gNw, B, 0, stream>>>(ht, row, start, eidx, normE, out, 1);
}